// AS_encoder_81853486728025
// MI455X (gfx1250) — compile-verified
//
#include <hip/hip_runtime.h>
#include <math.h>

#define DIM 64
#define SNEI 8

typedef __attribute__((ext_vector_type(2))) float v2f;
typedef __attribute__((ext_vector_type(8))) float v8f;

__device__ __forceinline__ float wave_reduce_add(float v) {
#pragma unroll
  for (int m = 16; m >= 1; m >>= 1) v += __shfl_xor(v, m, 32);
  return v;
}

// ---------------------------------------------------------------------------
// Y[n, j] = sum_k X[n,k] * W[j,k] + bias[j]      (X:[nrows,64], W:[64,64])
// MODE 0: store Y.   MODE 1: colsum[j] += sum_n tanh(Y[n,j])  (no Y store)
// One wave computes a 16x64 tile via 4 accumulators of V_WMMA_F32_16X16X4_F32.
// A 16x4 layout: lanes 0-15 rows M=0..15, VGPR0=K0/K2 (half), VGPR1=K1/K3.
// B 4x16 layout mirrored: lane half selects K0/K2, element selects +0/+1.
// ---------------------------------------------------------------------------
template <int MODE>
__global__ __launch_bounds__(128) void gemm64_wmma(
    const float* __restrict__ X, const float* __restrict__ W,
    const float* __restrict__ bias, float* __restrict__ Y,
    float* __restrict__ colsum, int nrows) {
  __shared__ float scol[DIM];
  if (MODE == 1) {
    if (threadIdx.x < DIM) scol[threadIdx.x] = 0.0f;
    __syncthreads();
  }

  const int wave = threadIdx.x >> 5;
  const int lane = threadIdx.x & 31;
  const int half = lane >> 4;   // 0: K+0/K+1, 1: K+2/K+3
  const int l    = lane & 15;
  const int row0 = blockIdx.x * 64 + wave * 16;

  v8f acc[4] = {};
  const int  rowA  = row0 + l;
  const bool rvalid = rowA < nrows;
  const float* xrow = X + (size_t)(rvalid ? rowA : 0) * DIM;
  const float* wl   = W + (size_t)l * DIM;   // row j = t*16 + l of W

#pragma unroll
  for (int kk = 0; kk < 16; ++kk) {
    const int k0 = kk * 4 + half * 2;
    float2 av = *(const float2*)(xrow + k0);
    v2f a;
    a.x = rvalid ? av.x : 0.0f;
    a.y = rvalid ? av.y : 0.0f;
#pragma unroll
    for (int t = 0; t < 4; ++t) {
      float2 bv = *(const float2*)(wl + t * 16 * DIM + k0);  // W[(t*16+l), k0..k0+1]
      v2f b; b.x = bv.x; b.y = bv.y;
      acc[t] = __builtin_amdgcn_wmma_f32_16x16x4_f32(
          false, a, false, b, (short)0, acc[t], false, false);
    }
  }

  // C layout: VGPR v, lanes 0-15 -> (M=v, N=lane); lanes 16-31 -> (M=v+8, N=lane-16)
#pragma unroll
  for (int t = 0; t < 4; ++t) {
    const int col = t * 16 + l;
    const float bj = bias[col];
    if (MODE == 0) {
#pragma unroll
      for (int v = 0; v < 8; ++v) {
        const int row = row0 + v + half * 8;
        if (row < nrows) Y[(size_t)row * DIM + col] = acc[t][v] + bj;
      }
    } else {
      float s = 0.0f;
#pragma unroll
      for (int v = 0; v < 8; ++v) {
        const int row = row0 + v + half * 8;
        if (row < nrows) s += tanhf(acc[t][v] + bj);
      }
      s += __shfl_xor(s, 16, 32);          // merge the two row-halves (same col)
      if (half == 0) atomicAdd(&scol[col], s);
    }
  }
  if (MODE == 1) {
    __syncthreads();
    if (threadIdx.x < DIM) atomicAdd(&colsum[threadIdx.x], scol[threadIdx.x]);
  }
}

// ---------------------------------------------------------------------------
// One wave per row n:
//   pa = dot(refp[n], att[0:64]);   ra = sum|refp[n]|
//   per neighbor s: na = dot(g_s, att[64:128]); nn = sum|g_s|
//   score_s = leaky_relu((pa+na)/max(ra+nn,1e-12)); softmax over S;
//   e[n] = sum_s w_s * g_s.      (eout may alias refp: row-exclusive)
// ---------------------------------------------------------------------------
__global__ __launch_bounds__(256) void attn_branch(
    const float* __restrict__ refp, const float* __restrict__ table,
    const int* __restrict__ nei, const float* __restrict__ att,
    float* __restrict__ eout, int nrows) {
  const int wave = threadIdx.x >> 5;
  const int lane = threadIdx.x & 31;
  const int row  = blockIdx.x * 8 + wave;
  if (row >= nrows) return;
  const int c2 = lane * 2;

  const float a0 = att[c2], a1 = att[c2 + 1];
  const float a2 = att[DIM + c2], a3 = att[DIM + c2 + 1];

  float2 p = *(const float2*)(refp + (size_t)row * DIM + c2);
  float pa = wave_reduce_add(p.x * a0 + p.y * a1);
  float ra = wave_reduce_add(fabsf(p.x) + fabsf(p.y));

  float2 g[SNEI];
  float  sc[SNEI];
#pragma unroll
  for (int s = 0; s < SNEI; ++s) {
    const int idx = nei[(size_t)row * SNEI + s];
    g[s] = *(const float2*)(table + (size_t)idx * DIM + c2);
    float na = wave_reduce_add(g[s].x * a2 + g[s].y * a3);
    float nn = wave_reduce_add(fabsf(g[s].x) + fabsf(g[s].y));
    float x  = (pa + na) / fmaxf(ra + nn, 1e-12f);
    sc[s] = (x >= 0.0f) ? x : 0.01f * x;   // leaky_relu(., 0.01)
  }
  float mx = sc[0];
#pragma unroll
  for (int s = 1; s < SNEI; ++s) mx = fmaxf(mx, sc[s]);
  float wsum = 0.0f, w[SNEI];
#pragma unroll
  for (int s = 0; s < SNEI; ++s) { w[s] = expf(sc[s] - mx); wsum += w[s]; }
  const float inv = 1.0f / wsum;
  float2 out; out.x = 0.0f; out.y = 0.0f;
#pragma unroll
  for (int s = 0; s < SNEI; ++s) {
    const float ws = w[s] * inv;
    out.x += ws * g[s].x;
    out.y += ws * g[s].y;
  }
  *(float2*)(eout + (size_t)row * DIM + c2) = out;
}

// beta[k] = softmax_k( dot(attvec, sp_k) * invN ), sp = K contiguous [64] sums
__global__ void beta_softmax_kernel(const float* __restrict__ attvec,
                                    const float* __restrict__ sp, int K,
                                    float invN, float* __restrict__ beta) {
  __shared__ float red[DIM];
  __shared__ float dots[4];
  const int t = threadIdx.x;
  for (int k = 0; k < K; ++k) {
    red[t] = attvec[t] * sp[k * DIM + t] * invN;
    __syncthreads();
    if (t == 0) {
      float s = 0.0f;
      for (int i = 0; i < DIM; ++i) s += red[i];
      dots[k] = s;
    }
    __syncthreads();
  }
  if (t == 0) {
    float m = dots[0];
    for (int k = 1; k < K; ++k) m = fmaxf(m, dots[k]);
    float s = 0.0f;
    for (int k = 0; k < K; ++k) { float e = expf(dots[k] - m); beta[k] = e; s += e; }
    for (int k = 0; k < K; ++k) beta[k] /= s;
  }
}

__global__ void combine2_elu_kernel(const float* __restrict__ eA,
                                    const float* __restrict__ eB,
                                    const float* __restrict__ beta,
                                    float* __restrict__ out, int n) {
  const int i = blockIdx.x * blockDim.x + threadIdx.x;
  if (i < n) {
    const float x = beta[0] * eA[i] + beta[1] * eB[i];
    out[i] = (x > 0.0f) ? x : (expf(x) - 1.0f);   // elu, alpha=1
  }
}

__global__ void combine3_kernel(const float* __restrict__ e0,
                                const float* __restrict__ e1,
                                const float* __restrict__ tf,
                                const float* __restrict__ beta,
                                float* __restrict__ out, int n) {
  const int i = blockIdx.x * blockDim.x + threadIdx.x;
  if (i < n) out[i] = beta[0] * e0[i] + beta[1] * e1[i] + beta[2] * tf[i];
}

__global__ void zero_kernel(float* __restrict__ p, int n) {
  const int i = blockIdx.x * blockDim.x + threadIdx.x;
  if (i < n) p[i] = 0.0f;
}

// ---------------------------------------------------------------------------
extern "C" void kernel_launch(void* const* d_in, const int* in_sizes, int n_in,
                              void* d_out, int out_size, void* d_ws, size_t ws_size,
                              hipStream_t stream) {
  const float* tf  = (const float*)d_in[0];
  const float* h0  = (const float*)d_in[1];
  const float* h1  = (const float*)d_in[2];
  const float* h2  = (const float*)d_in[3];
  const float* rf0 = (const float*)d_in[4];
  const float* rf1 = (const float*)d_in[5];
  const float* rf2 = (const float*)d_in[6];
  const int*   nei0 = (const int*)d_in[7];
  const int*   nei1 = (const int*)d_in[8];
  // per-relation params start at 9 and 18
  const float* interW   = (const float*)d_in[27];
  const float* interb   = (const float*)d_in[28];
  const float* interatt = (const float*)d_in[29];

  const int N  = in_sizes[0] / DIM;
  const int NE = N * DIM;

  float* ws   = (float*)d_ws;
  float* bufA = ws;            // refp_attr, then e_attr in place
  float* bufB = ws + (size_t)NE;
  float* e0b  = ws + 2 * (size_t)NE;
  float* e1b  = ws + 3 * (size_t)NE;
  float* sp   = ws + 4 * (size_t)NE;  // 7 x [64] column-sum accumulators
  float* betas = sp + 7 * DIM;        // [2]+[2]+[3]

  const int gemmBlocks = (N + 63) / 64;
  const int attnBlocks = (N + 7) / 8;
  const int elemBlocks = (NE + 255) / 256;
  const float invN = 1.0f / (float)N;

  zero_kernel<<<2, 256, 0, stream>>>(sp, 7 * DIM + 7);

  for (int r = 0; r < 2; ++r) {
    const int p = 9 + 9 * r;
    const float* att    = (const float*)d_in[p + 0];
    const float* att2   = (const float*)d_in[p + 1];
    const float* Wr     = (const float*)d_in[p + 2];
    const float* br     = (const float*)d_in[p + 3];
    const float* Wr2    = (const float*)d_in[p + 4];
    const float* br2    = (const float*)d_in[p + 5];
    const float* aggW   = (const float*)d_in[p + 6];
    const float* aggb   = (const float*)d_in[p + 7];
    const float* aggatt = (const float*)d_in[p + 8];
    const float* tab    = (r == 0) ? h1 : h2;
    const float* rtab   = (r == 0) ? rf1 : rf2;
    const int*   nei    = (r == 0) ? nei0 : nei1;
    float* spR   = sp + 2 * r * DIM;     // [2][64]
    float* betaR = betas + 2 * r;        // [2]
    float* eR    = (r == 0) ? e0b : e1b;

    // refp projections (WMMA f32 GEMM + bias)
    gemm64_wmma<0><<<gemmBlocks, 128, 0, stream>>>(h0,  Wr,  br,  bufA, nullptr, N);
    gemm64_wmma<0><<<gemmBlocks, 128, 0, stream>>>(rf0, Wr2, br2, bufB, nullptr, N);
    // attention branches (in-place: e overwrites refp row-by-row)
    attn_branch<<<attnBlocks, 256, 0, stream>>>(bufA, tab,  nei, att,  bufA, N);
    attn_branch<<<attnBlocks, 256, 0, stream>>>(bufB, rtab, nei, att2, bufB, N);
    // sp(e) column sums of tanh(e @ aggW^T + aggb)   (WMMA + fused reduce)
    gemm64_wmma<1><<<gemmBlocks, 128, 0, stream>>>(bufA, aggW, aggb, nullptr, spR,        N);
    gemm64_wmma<1><<<gemmBlocks, 128, 0, stream>>>(bufB, aggW, aggb, nullptr, spR + DIM,  N);
    beta_softmax_kernel<<<1, DIM, 0, stream>>>(aggatt, spR, 2, invN, betaR);
    combine2_elu_kernel<<<elemBlocks, 256, 0, stream>>>(bufA, bufB, betaR, eR, NE);
  }

  // inter-relation semantic aggregation
  float* sps   = sp + 4 * DIM;   // [3][64]
  float* betaF = betas + 4;      // [3]
  gemm64_wmma<1><<<gemmBlocks, 128, 0, stream>>>(e0b, interW, interb, nullptr, sps,           N);
  gemm64_wmma<1><<<gemmBlocks, 128, 0, stream>>>(e1b, interW, interb, nullptr, sps + DIM,     N);
  gemm64_wmma<1><<<gemmBlocks, 128, 0, stream>>>(tf,  interW, interb, nullptr, sps + 2 * DIM, N);
  beta_softmax_kernel<<<1, DIM, 0, stream>>>(interatt, sps, 3, invN, betaF);
  combine3_kernel<<<elemBlocks, 256, 0, stream>>>(e0b, e1b, tf, betaF, (float*)d_out, NE);
}